// TripletLoss_32796370273130
// MI455X (gfx1250) — compile-verified
//
#include <hip/hip_runtime.h>
#include <hip/hip_bf16.h>

typedef __attribute__((ext_vector_type(2))) float v2f;
typedef __attribute__((ext_vector_type(8))) float v8f;

#define N 8192
#define D 128
#define TILE 16
#define NTILE 2                          // column tiles processed per iteration
#define CSTEP (TILE * NTILE)             // 32 columns per iteration
#define WAVES 8
#define ROWS_PER_BLOCK (TILE * WAVES)    // 128
#define ROW_BLOCKS (N / ROWS_PER_BLOCK)  // 64
#define COL_SPLITS 8
#define COLS_PER_SPLIT (N / COL_SPLITS)  // 1024
#define MARGIN 0.3f
#define EPS 1e-12f

// ---------------- per-row squared norms ----------------
__global__ __launch_bounds__(256) void sqnorm_kernel(const float* __restrict__ X,
                                                     float* __restrict__ sq) {
    const int wave = threadIdx.x >> 5;
    const int lane = threadIdx.x & 31;
    const int row  = blockIdx.x * 8 + wave;
    const float4 v = ((const float4*)(X + (size_t)row * D))[lane];
    float s = v.x * v.x + v.y * v.y + v.z * v.z + v.w * v.w;
    #pragma unroll
    for (int off = 16; off; off >>= 1) s += __shfl_xor(s, off, 32);
    if (lane == 0) sq[row] = s;
}

// ---------------- workspace init (ap=0 bits, an=+inf bits) ----------------
__global__ __launch_bounds__(256) void init_ws_kernel(unsigned* __restrict__ apBits,
                                                      unsigned* __restrict__ anBits) {
    const int i = blockIdx.x * 256 + threadIdx.x;
    if (i < N) {
        apBits[i] = 0u;          // 0.0f  (all dists > 0)
        anBits[i] = 0x7f800000u; // +inf
    }
}

// ---------------- WMMA batch-hard mining ----------------
__global__ __launch_bounds__(256) void hardmine_kernel(const float* __restrict__ X,
                                                       const int* __restrict__ T,
                                                       const float* __restrict__ sq,
                                                       unsigned* __restrict__ apBits,
                                                       unsigned* __restrict__ anBits) {
    __shared__ float Bs[CSTEP][D + 4]; // stride 132 floats -> conflict-free (bank = 4*col + k mod 64)
    __shared__ int   Ts[CSTEP];
    __shared__ float Ss[CSTEP];

    const int tid  = threadIdx.x;
    const int wave = tid >> 5;
    const int lane = tid & 31;
    const int l15  = lane & 15;
    const bool hi  = lane >= 16;

    const int rowBase = blockIdx.x * ROWS_PER_BLOCK + wave * TILE;
    const int myRow   = rowBase + l15;

    // --- preload A operand registers: ISA 16x4 f32 A layout per k-step ---
    // lanes 0-15: row M=l15, K = 4s+0 / 4s+1 ; lanes 16-31: row M=l15, K = 4s+2 / 4s+3
    v2f a[32];
    {
        const float* xr = X + (size_t)myRow * D + (hi ? 2 : 0);
        #pragma unroll
        for (int s = 0; s < 32; ++s) {
            a[s].x = xr[4 * s];
            a[s].y = xr[4 * s + 1];
        }
    }
    // labels / norms of the 8 C-matrix rows this lane covers (M = k + (hi?8:0))
    int   tr[8];
    float sr[8];
    #pragma unroll
    for (int k = 0; k < 8; ++k) {
        const int m = rowBase + k + (hi ? 8 : 0);
        tr[k] = T[m];
        sr[k] = sq[m];
    }

    float vap[8], van[8];
    #pragma unroll
    for (int k = 0; k < 8; ++k) { vap[k] = 0.0f; van[k] = __uint_as_float(0x7f800000u); }

    const int colStart = blockIdx.y * COLS_PER_SPLIT;

    for (int ct = 0; ct < COLS_PER_SPLIT / CSTEP; ++ct) {
        const int colBase = colStart + ct * CSTEP;

        __syncthreads(); // protect previous tile's consumers
        // stage 32x128 B panel (rows colBase..colBase+31 of X) into LDS
        {
            const float* src = X + (size_t)colBase * D;
            #pragma unroll
            for (int i = tid; i < CSTEP * D; i += 256) {
                const int c = i >> 7;    // / 128
                const int k = i & 127;
                Bs[c][k] = src[i];
            }
            if (tid < CSTEP) {
                Ts[tid] = T[colBase + tid];
                Ss[tid] = sq[colBase + tid];
            }
        }
        __syncthreads();

        // --- two 16x16 Gram tiles: two independent WMMA chains sharing A ---
        v8f acc0 = {};
        v8f acc1 = {};
        #pragma unroll
        for (int s = 0; s < 32; ++s) {
            // B 4x16 layout: lanes 0-15: K=4s+0/4s+1 at N=lane; lanes 16-31: K=4s+2/4s+3
            const int bk = (s << 2) + (hi ? 2 : 0);
            v2f b0 = *(const v2f*)&Bs[l15][bk];
            v2f b1 = *(const v2f*)&Bs[TILE + l15][bk];
            acc0 = __builtin_amdgcn_wmma_f32_16x16x4_f32(
                false, a[s], false, b0, (short)0, acc0, false, false);
            acc1 = __builtin_amdgcn_wmma_f32_16x16x4_f32(
                false, a[s], false, b1, (short)0, acc1, false, false);
        }

        // --- epilogue: distances + hard mining, matching C VGPR layout ---
        const float sc0 = Ss[l15];
        const int   tc0 = Ts[l15];
        const float sc1 = Ss[TILE + l15];
        const int   tc1 = Ts[TILE + l15];
        #pragma unroll
        for (int k = 0; k < 8; ++k) {
            const float d20 = sr[k] + sc0 - 2.0f * acc0[k];
            const float d21 = sr[k] + sc1 - 2.0f * acc1[k];
            const float dist0 = sqrtf(fmaxf(d20, EPS));
            const float dist1 = sqrtf(fmaxf(d21, EPS));
            if (tr[k] == tc0) vap[k] = fmaxf(vap[k], dist0);
            else              van[k] = fminf(van[k], dist0);
            if (tr[k] == tc1) vap[k] = fmaxf(vap[k], dist1);
            else              van[k] = fminf(van[k], dist1);
        }
    }

    // butterfly reduce across each 16-lane half (columns of the tile)
    #pragma unroll
    for (int off = 1; off < 16; off <<= 1) {
        #pragma unroll
        for (int k = 0; k < 8; ++k) {
            vap[k] = fmaxf(vap[k], __shfl_xor(vap[k], off, 32));
            van[k] = fminf(van[k], __shfl_xor(van[k], off, 32));
        }
    }
    if (l15 == 0) { // lanes 0 and 16 hold reduced rows M=k and M=8+k
        const int mbase = rowBase + (hi ? 8 : 0);
        #pragma unroll
        for (int k = 0; k < 8; ++k) {
            atomicMax(&apBits[mbase + k], __float_as_uint(vap[k]));
            atomicMin(&anBits[mbase + k], __float_as_uint(van[k]));
        }
    }
}

// ---------------- final loss / precision reduction ----------------
__global__ __launch_bounds__(1024) void finalize_kernel(const unsigned* __restrict__ apBits,
                                                        const unsigned* __restrict__ anBits,
                                                        float* __restrict__ out) {
    __shared__ float sl[32], sp[32];
    float lsum = 0.0f, psum = 0.0f;
    for (int i = threadIdx.x; i < N; i += 1024) {
        const float ap = __uint_as_float(apBits[i]);
        const float an = __uint_as_float(anBits[i]);
        lsum += fmaxf(ap - an + MARGIN, 0.0f);
        psum += (an > ap) ? 1.0f : 0.0f;
    }
    #pragma unroll
    for (int off = 16; off; off >>= 1) {
        lsum += __shfl_xor(lsum, off, 32);
        psum += __shfl_xor(psum, off, 32);
    }
    const int wave = threadIdx.x >> 5, lane = threadIdx.x & 31;
    if (lane == 0) { sl[wave] = lsum; sp[wave] = psum; }
    __syncthreads();
    if (wave == 0) {
        lsum = sl[lane];
        psum = sp[lane];
        #pragma unroll
        for (int off = 16; off; off >>= 1) {
            lsum += __shfl_xor(lsum, off, 32);
            psum += __shfl_xor(psum, off, 32);
        }
        if (lane == 0) {
            out[0] = lsum / (float)N; // loss
            out[1] = psum / (float)N; // prec
        }
    }
}

extern "C" void kernel_launch(void* const* d_in, const int* in_sizes, int n_in,
                              void* d_out, int out_size, void* d_ws, size_t ws_size,
                              hipStream_t stream) {
    const float* X = (const float*)d_in[0];
    const int*   T = (const int*)d_in[1];

    float*    sq     = (float*)d_ws;
    unsigned* apBits = (unsigned*)((char*)d_ws + (size_t)N * sizeof(float));
    unsigned* anBits = apBits + N;
    float*    out    = (float*)d_out;

    sqnorm_kernel<<<N / 8, 256, 0, stream>>>(X, sq);
    init_ws_kernel<<<(N + 255) / 256, 256, 0, stream>>>(apBits, anBits);
    hardmine_kernel<<<dim3(ROW_BLOCKS, COL_SPLITS), 256, 0, stream>>>(X, T, sq, apBits, anBits);
    finalize_kernel<<<1, 1024, 0, stream>>>(apBits, anBits, out);
}